// BoundaryLoss_50534585204952
// MI455X (gfx1250) — compile-verified
//
#include <hip/hip_runtime.h>
#include <hip/hip_bf16.h>

typedef __attribute__((ext_vector_type(2))) float v2f;
typedef __attribute__((ext_vector_type(8))) float v8f;

#define Bdim 8
#define Cdim 4
#define Hdim 256
#define Wdim 256

__device__ __forceinline__ int imin(int a, int b) { return a < b ? a : b; }

// ---------------------------------------------------------------------------
// Kernel A: exact min-plus pass along W for all 4 classes, both masks.
// One block per (b,h) line. Output: squared distance along the row to the
// nearest pixel with y==c (pos) / y!=c (neg), as u16 (0xFFFF = none in row).
// ---------------------------------------------------------------------------
__global__ void __launch_bounds__(256) edt_row_pass(const int* __restrict__ y,
                                                    unsigned short* __restrict__ g1pos,
                                                    unsigned short* __restrict__ g1neg) {
  const int blk = blockIdx.x;          // b*H + h
  const int b = blk >> 8;
  const int h = blk & 255;
  const int i = threadIdx.x;

  __shared__ int yrow[256];
  yrow[i] = y[(b * Hdim + h) * Wdim + i];
  __syncthreads();

  const int BIGI = 1 << 20;            // > max d2 (65025), safe in adds
  int bp0 = BIGI, bp1 = BIGI, bp2 = BIGI, bp3 = BIGI;
  int bn0 = BIGI, bn1 = BIGI, bn2 = BIGI, bn3 = BIGI;

#pragma unroll 4
  for (int j = 0; j < 256; ++j) {
    const int cls = yrow[j];
    const int dd = i - j;
    const int d2 = dd * dd;
    bp0 = imin(bp0, (cls == 0) ? d2 : BIGI);
    bn0 = imin(bn0, (cls == 0) ? BIGI : d2);
    bp1 = imin(bp1, (cls == 1) ? d2 : BIGI);
    bn1 = imin(bn1, (cls == 1) ? BIGI : d2);
    bp2 = imin(bp2, (cls == 2) ? d2 : BIGI);
    bn2 = imin(bn2, (cls == 2) ? BIGI : d2);
    bp3 = imin(bp3, (cls == 3) ? d2 : BIGI);
    bn3 = imin(bn3, (cls == 3) ? BIGI : d2);
  }

  const size_t cstr = (size_t)Hdim * Wdim;
  const size_t base = ((size_t)(b * Cdim) * Hdim + h) * Wdim + i;
  g1pos[base + 0 * cstr] = (bp0 >= BIGI) ? 0xFFFFu : (unsigned short)bp0;
  g1pos[base + 1 * cstr] = (bp1 >= BIGI) ? 0xFFFFu : (unsigned short)bp1;
  g1pos[base + 2 * cstr] = (bp2 >= BIGI) ? 0xFFFFu : (unsigned short)bp2;
  g1pos[base + 3 * cstr] = (bp3 >= BIGI) ? 0xFFFFu : (unsigned short)bp3;
  g1neg[base + 0 * cstr] = (bn0 >= BIGI) ? 0xFFFFu : (unsigned short)bn0;
  g1neg[base + 1 * cstr] = (bn1 >= BIGI) ? 0xFFFFu : (unsigned short)bn1;
  g1neg[base + 2 * cstr] = (bn2 >= BIGI) ? 0xFFFFu : (unsigned short)bn2;
  g1neg[base + 3 * cstr] = (bn3 >= BIGI) ? 0xFFFFu : (unsigned short)bn3;
}

// ---------------------------------------------------------------------------
// Kernel B: min-plus pass along H + softmax + boundary map + partial sums.
// One block per (b, w) column; thread = h. Partials written in fixed order.
// ---------------------------------------------------------------------------
__global__ void __launch_bounds__(256) edt_col_loss(const float* __restrict__ x,
                                                    const int* __restrict__ y,
                                                    const unsigned short* __restrict__ g1pos,
                                                    const unsigned short* __restrict__ g1neg,
                                                    float* __restrict__ partial) {
  const int blk = blockIdx.x;          // b*W + i
  const int b = blk >> 8;
  const int i = blk & 255;
  const int h = threadIdx.x;

  __shared__ int colPos[4 * 256];
  __shared__ int colNeg[4 * 256];

  const size_t cstr = (size_t)Hdim * Wdim;
  {
    const size_t base = ((size_t)(b * Cdim) * Hdim + h) * Wdim + i;
#pragma unroll
    for (int c = 0; c < 4; ++c) {
      const unsigned up = g1pos[base + (size_t)c * cstr];
      const unsigned un = g1neg[base + (size_t)c * cstr];
      colPos[c * 256 + h] = (up == 0xFFFFu) ? (1 << 29) : (int)up;
      colNeg[c * 256 + h] = (un == 0xFFFFu) ? (1 << 29) : (int)un;
    }
  }
  __syncthreads();

  int dn2[4], dp2[4];
#pragma unroll
  for (int c = 0; c < 4; ++c) {
    int bp = 0x7F000000, bn = 0x7F000000;
    const int* cp = &colPos[c * 256];
    const int* cn = &colNeg[c * 256];
#pragma unroll 4
    for (int k = 0; k < 256; ++k) {
      const int dd = h - k;
      const int d2 = dd * dd;
      bp = imin(bp, d2 + cp[k]);
      bn = imin(bn, d2 + cn[k]);
    }
    dn2[c] = bp;   // EDT(posmask)^2 -> dist to nearest positive pixel
    dp2[c] = bn;   // EDT(negmask)^2 -> dist to nearest negative pixel
  }

  const int ylab = y[(b * Hdim + h) * Wdim + i];

  float xs[4];
#pragma unroll
  for (int c = 0; c < 4; ++c)
    xs[c] = x[((size_t)(b * Cdim + c) * Hdim + h) * Wdim + i];

  float mx = fmaxf(fmaxf(xs[0], xs[1]), fmaxf(xs[2], xs[3]));
  float e[4];
  float sum = 0.0f;
#pragma unroll
  for (int c = 0; c < 4; ++c) {
    e[c] = expf(xs[c] - mx);
    sum += e[c];
  }
  const float inv = 1.0f / sum;

  float acc = 0.0f;
#pragma unroll
  for (int c = 0; c < 4; ++c) {
    const float dn = sqrtf((float)dn2[c]);
    const float dp = sqrtf((float)dp2[c]);
    const float bd = (ylab == c) ? -dp : dn;   // bdistmap at this pixel
    acc += (e[c] * inv) * bd;
  }

  // deterministic block tree reduction
  __shared__ float red[256];
  red[h] = acc;
  __syncthreads();
  for (int s = 128; s > 0; s >>= 1) {
    if (h < s) red[h] += red[h + s];
    __syncthreads();
  }
  if (h == 0) partial[blk] = red[0];
}

// ---------------------------------------------------------------------------
// Kernel C: sum 2048 partials with V_WMMA_F32_16X16X4_F32 (B = all-ones),
// one wave, f32-exact, deterministic. 32 tiles of 64 values each.
// ---------------------------------------------------------------------------
__global__ void __launch_bounds__(32) final_reduce_wmma(const float* __restrict__ partial,
                                                        float* __restrict__ out) {
  const int lane = threadIdx.x;        // 32 threads = 1 wave, EXEC all ones
  v8f acc = {0.f, 0.f, 0.f, 0.f, 0.f, 0.f, 0.f, 0.f};
  v2f ones;
  ones.x = 1.0f;
  ones.y = 1.0f;

  for (int t = 0; t < 32; ++t) {
    v2f a;
    a.x = partial[t * 64 + lane];
    a.y = partial[t * 64 + 32 + lane];
    // D = A(16x4) * ones(4x16) + C : every D column accumulates row-sums of A
    acc = __builtin_amdgcn_wmma_f32_16x16x4_f32(
        /*neg_a=*/false, a, /*neg_b=*/false, ones,
        /*c_mod=*/(short)0, acc, /*reuse_a=*/false, /*reuse_b=*/false);
  }

  // column n of D: rows 0..7 in lane n, rows 8..15 in lane n+16
  float s = acc[0] + acc[1] + acc[2] + acc[3] + acc[4] + acc[5] + acc[6] + acc[7];
  __shared__ float sh[2];
  if (lane == 0) sh[0] = s;
  if (lane == 16) sh[1] = s;
  __syncthreads();
  if (lane == 0)
    out[0] = (sh[0] + sh[1]) * (1.0f / (float)(Bdim * Cdim * Hdim * Wdim));
}

// ---------------------------------------------------------------------------
extern "C" void kernel_launch(void* const* d_in, const int* in_sizes, int n_in,
                              void* d_out, int out_size, void* d_ws, size_t ws_size,
                              hipStream_t stream) {
  const float* x = (const float*)d_in[0];
  const int* y = (const int*)d_in[1];
  float* out = (float*)d_out;

  const size_t mapElems = (size_t)Bdim * Cdim * Hdim * Wdim;   // 2,097,152
  unsigned short* g1pos = (unsigned short*)d_ws;
  unsigned short* g1neg = (unsigned short*)((char*)d_ws + mapElems * sizeof(unsigned short));
  float* partial = (float*)((char*)d_ws + 2 * mapElems * sizeof(unsigned short));

  edt_row_pass<<<Bdim * Hdim, 256, 0, stream>>>(y, g1pos, g1neg);
  edt_col_loss<<<Bdim * Wdim, 256, 0, stream>>>(x, y, g1pos, g1neg, partial);
  final_reduce_wmma<<<1, 32, 0, stream>>>(partial, out);
}